// GKernel_90469191123195
// MI455X (gfx1250) — compile-verified
//
#include <hip/hip_runtime.h>
#include <math.h>

// ---------------------------------------------------------------------------
// Rotated-steerable weight generation for MI455X (gfx1250).
//
// out[ci, o, co, i, py, px] (128,8,128,8,5,5) f32, 104.9 MB -> bandwidth bound
// (write floor ~4.5us @ 23.3 TB/s). The angular-lerp + bilinear-resample +
// mask is expressed per (o,i) pair as a GEMM:
//     out[f, p] = sum_s A[f, s] * B[s, p]
//   A : 16384 x 52   (rows = (co,ci) channel pairs; cols = {i0 slab, i1 slab})
//   B : 52 x 32      (per-pair coefficient matrix, <=8 nonzeros per column,
//                     folds angular lerp x bilinear weights x disc mask)
// computed with V_WMMA_F32_16X16X4_F32 (full f32 precision).
// ---------------------------------------------------------------------------

typedef float v2f __attribute__((ext_vector_type(2)));
typedef float v8f __attribute__((ext_vector_type(8)));

#define TWO_PI_F 6.2831853071795864769f

__launch_bounds__(128, 1)
__global__ void rot_weight_wmma(const float* __restrict__ in_H,
                                const float* __restrict__ out_H,
                                const float* __restrict__ weight,
                                const float* __restrict__ grid_Rn,
                                const float* __restrict__ mask,
                                float* __restrict__ out)
{
    // B stored transposed: sB[n][k], n = output tap (0..31), k = source (0..51).
    // Stride 52 -> 8B aligned v2f fragment loads, bank-conflict free (52r mod 64
    // distinct for r=0..15, and the +2 offset of lanes 16-31 cannot collide).
    __shared__ float sB[32 * 52];
    __shared__ float sA[4][16 * 52];   // per-wave A staging tile (16 rows x 52)

    const int tid  = threadIdx.x;
    const int lane = tid & 31;
    const int wv   = tid >> 5;

    const int m = blockIdx.y;          // (o,i) pair
    const int o = m >> 3;
    const int i = m & 7;

    // ---- angular interpolation parameters (every thread; needed for A bases)
    const float oh = out_H[o];
    const float ih = in_H[i];
    float theta = ih - oh;
    float rmod  = fmodf(theta, TWO_PI_F);
    if (rmod < 0.0f) rmod += TWO_PI_F;             // jnp.mod semantics
    const float pos  = rmod * (8.0f / TWO_PI_F);
    const float posf = floorf(pos);
    const int   i0   = ((int)posf) & 7;
    const int   i1   = (i0 + 1) & 7;
    const float fr   = pos - posf;

    // ---- zero + build coefficient matrix B (once per block) ----------------
    for (int idx = tid; idx < 32 * 52; idx += 128) sB[idx] = 0.0f;
    __syncthreads();

    if (tid < 25) {
        const int p = tid;                         // p = py*5 + px
        const float X  = grid_Rn[p];
        const float Y  = grid_Rn[25 + p];
        const float mk = mask[p];
        const float c = cosf(-oh);
        const float s = sinf(-oh);
        const float gx = c * X - s * Y;
        const float gy = s * X + c * Y;
        const float x = (gx + 1.0f) * 2.0f;        // (g+1)*0.5*(5-1)
        const float y = (gy + 1.0f) * 2.0f;
        const float x0 = floorf(x);
        const float y0 = floorf(y);
        const float fx = x - x0;
        const float fy = y - y0;
        int x0i = (int)x0; x0i = x0i < 0 ? 0 : (x0i > 4 ? 4 : x0i);
        int x1i = x0i + 1; x1i = x1i > 4 ? 4 : x1i;
        int y0i = (int)y0; y0i = y0i < 0 ? 0 : (y0i > 4 ? 4 : y0i);
        int y1i = y0i + 1; y1i = y1i > 4 ? 4 : y1i;
        const float a0 = (1.0f - fr) * mk;         // mask folded into B
        const float a1 = fr * mk;
        const float w00 = (1.0f - fy) * (1.0f - fx);
        const float w01 = (1.0f - fy) * fx;
        const float w10 = fy * (1.0f - fx);
        const float w11 = fy * fx;
        float* col = &sB[p * 52];
        const int q00 = y0i * 5 + x0i;
        const int q01 = y0i * 5 + x1i;
        const int q10 = y1i * 5 + x0i;
        const int q11 = y1i * 5 + x1i;
        col[q00] += a0 * w00; col[25 + q00] += a1 * w00;
        col[q01] += a0 * w01; col[25 + q01] += a1 * w01;
        col[q10] += a0 * w10; col[25 + q10] += a1 * w10;
        col[q11] += a0 * w11; col[25 + q11] += a1 * w11;
    }
    __syncthreads();

    float* sAw = sA[wv];
    const int rrow = lane & 15;        // M row / N col within fragment
    const int hi   = lane >> 4;        // lane-half selects K pair / M half
    const int hi2  = hi << 1;

    // 1024 row-tiles total = gridDim.x(32) * 4 waves * 8 tiles
    for (int tt = 0; tt < 8; ++tt) {
        const int tile = blockIdx.x * 32 + wv * 8 + tt;
        const int f0   = tile << 4;            // first channel-flat row
        const int co   = f0 >> 7;
        const int ci0  = f0 & 127;

        // A rows are two contiguous 400-float slabs of weight (coalesced)
        const float* g0 = weight + (((size_t)i0 * 128 + co) * 128 + ci0) * 25;
        const float* g1 = weight + (((size_t)i1 * 128 + co) * 128 + ci0) * 25;

        for (int idx = lane; idx < 400; idx += 32) {
            const int rr = idx / 25;
            const int cc = idx - rr * 25;
            sAw[rr * 52 + cc]      = g0[idx];
            sAw[rr * 52 + cc + 25] = g1[idx];
        }
        // zero K-padding columns 50,51 (16 rows x 2 cols by 32 lanes)
        sAw[(lane >> 1) * 52 + 50 + (lane & 1)] = 0.0f;
        asm volatile("s_wait_dscnt 0" ::: "memory");

        v8f acc0 = {0.f,0.f,0.f,0.f,0.f,0.f,0.f,0.f};
        v8f acc1 = {0.f,0.f,0.f,0.f,0.f,0.f,0.f,0.f};

        #pragma unroll
        for (int k = 0; k < 13; ++k) {
            const int kb = k * 4 + hi2;        // K = 4k + 2*(lane/16) + v
            const v2f a  = *(const v2f*)&sAw[rrow * 52 + kb];
            const v2f b0 = *(const v2f*)&sB[rrow * 52 + kb];
            const v2f b1 = *(const v2f*)&sB[(16 + rrow) * 52 + kb];
            acc0 = __builtin_amdgcn_wmma_f32_16x16x4_f32(
                       false, a, false, b0, (short)0, acc0, false, false);
            acc1 = __builtin_amdgcn_wmma_f32_16x16x4_f32(
                       false, a, false, b1, (short)0, acc1, false, false);
        }

        // D layout: VGPR r, lanes 0-15 -> row r, lanes 16-31 -> row r+8; col = lane%16
        #pragma unroll
        for (int rv = 0; rv < 8; ++rv) {
            const int ci = ci0 + rv + 8 * hi;
            const size_t base = ((((size_t)ci * 8 + o) * 128 + co) * 8 + i) * 25;
            out[base + rrow] = acc0[rv];               // taps 0..15
            if (rrow < 9) out[base + 16 + rrow] = acc1[rv];  // taps 16..24
        }
    }
}

extern "C" void kernel_launch(void* const* d_in, const int* in_sizes, int n_in,
                              void* d_out, int out_size, void* d_ws, size_t ws_size,
                              hipStream_t stream) {
    (void)in_sizes; (void)n_in; (void)out_size; (void)d_ws; (void)ws_size;
    const float* in_H    = (const float*)d_in[0];
    const float* out_H   = (const float*)d_in[1];
    const float* weight  = (const float*)d_in[2];
    // d_in[3] = grid_H : only its length (N=8) matters, baked into constants
    const float* grid_Rn = (const float*)d_in[4];
    const float* mask    = (const float*)d_in[5];
    float* out = (float*)d_out;

    dim3 grid(32, 64, 1);   // 32 tile-groups x 64 (o,i) pairs
    dim3 block(128, 1, 1);  // 4 wave32s
    hipLaunchKernelGGL(rot_weight_wmma, grid, block, 0, stream,
                       in_H, out_H, weight, grid_Rn, mask, out);
}